// HOGLayer_47012712022575
// MI455X (gfx1250) — compile-verified
//
#include <hip/hip_runtime.h>
#include <hip/hip_bf16.h>
#include <stdint.h>

#define NBINS 10
#define POOL  8
#define IMG   512
#define TILE  64
#define HALO  66         // TILE + 2 halo for 3x3 sobel
#define TPB   256        // 8 wave32 waves

// ---------------- CDNA5 async global->LDS helpers ----------------

typedef __attribute__((address_space(1))) int* as1_i32p;
typedef __attribute__((address_space(3))) int* as3_i32p;

__device__ __forceinline__ void async_g2l_b32(const float* g, float* l) {
#if defined(__has_builtin) && __has_builtin(__builtin_amdgcn_global_load_async_to_lds_b32)
    __builtin_amdgcn_global_load_async_to_lds_b32(
        (as1_i32p)(unsigned long long)(uintptr_t)g,
        (as3_i32p)(unsigned)(uintptr_t)l,
        0, 0);
#else
    unsigned           loff = (unsigned)(uintptr_t)l;
    unsigned long long ga   = (unsigned long long)(uintptr_t)g;
    asm volatile("global_load_async_to_lds_b32 %0, %1, off"
                 :: "v"(loff), "v"(ga) : "memory");
#endif
}

__device__ __forceinline__ void wait_asynccnt0() {
#if defined(__has_builtin) && __has_builtin(__builtin_amdgcn_s_wait_asynccnt)
    __builtin_amdgcn_s_wait_asynccnt(0);
#else
    asm volatile("s_wait_asynccnt 0" ::: "memory");
#endif
}

// LDS float atomic add: guaranteed single ds_add_f32 (no CAS loop)
__device__ __forceinline__ void ds_fadd(unsigned lds_byte_addr, float v) {
    asm volatile("ds_add_f32 %0, %1" :: "v"(lds_byte_addr), "v"(v) : "memory");
}

// ---------------- TDM (Tensor Data Mover) path: clang-23 6-arg builtin ----------------
#if defined(__has_builtin)
#if __has_builtin(__builtin_amdgcn_tensor_load_to_lds) && __has_builtin(__builtin_amdgcn_s_wait_tensorcnt)
#define HAVE_TDM 1
#endif
#endif

#ifdef HAVE_TDM
typedef unsigned int v4u __attribute__((ext_vector_type(4)));
typedef int          v8i __attribute__((ext_vector_type(8)));
typedef int          v4i __attribute__((ext_vector_type(4)));

// One-shot DMA of a 66x66 f32 tile (row stride 512 elements) from global to LDS.
__device__ __forceinline__ void tdm_load_tile(const float* gsrc, float* ldst) {
    unsigned long long ga = (unsigned long long)(uintptr_t)gsrc;
    unsigned lds_base = (unsigned)(uintptr_t)ldst;
    // D# group 0: count=1 (valid), lds_addr, global_addr[56:0], type=2 ("image")
    v4u g0 = { 1u,
               lds_base,
               (unsigned)(ga & 0xFFFFFFFFu),
               (unsigned)((ga >> 32) & 0x01FFFFFFu) | (2u << 30) };
    // D# group 1:
    //  d0: wg_mask=0, data_size=2 (4B), no flags
    //  d1: atomic_barrier_addr=0, tensor_dim0[15:0]=0       (tensor_dim0 = 1<<30, huge => no OOB)
    //  d2: tensor_dim0[31:16]=0x4000, tensor_dim1[15:0]=0   (tensor_dim1 = 1<<30)
    //  d3: tensor_dim1[31:16]=0x4000, tile_dim0=66
    //  d4: tile_dim1=66, tile_dim2=0
    //  d5: tensor_dim0_stride[31:0]=512
    //  d6: tensor_dim0_stride[47:32]=0, tensor_dim1_stride[15:0]=0
    //  d7: tensor_dim1_stride[47:16]=0
    v8i g1 = { 0x20000, 0, 0x4000, (HALO << 16) | 0x4000, HALO, IMG, 0, 0 };
    v4i g2 = { 0, 0, 0, 0 };
    v4i g3 = { 0, 0, 0, 0 };
    v8i g4 = { 0, 0, 0, 0, 0, 0, 0, 0 };   // unused trailing operand group (VADDR4)
    __builtin_amdgcn_tensor_load_to_lds(g0, g1, g2, g3, g4, 0);
}
#endif

// ---------------- kernel ----------------

__global__ __launch_bounds__(TPB)
void hog_fused_kernel(const float* __restrict__ x, float* __restrict__ out) {
    __shared__ float tile[HALO * HALO];   // 17424 B staged input tile (+halo)
    __shared__ float hist[64 * NBINS];    //  2560 B per-cell histograms

    const int tid = threadIdx.x;
    const int tX  = blockIdx.x;           // 0..7  tile col
    const int tY  = blockIdx.y;           // 0..7  tile row
    const int n   = blockIdx.z;           // batch

    const int row0 = tY * TILE;
    const int col0 = tX * TILE;
    const float* xin = x + (size_t)n * IMG * IMG;

    // interior tiles have their full 66x66 halo inside the image (uniform per block)
    const bool interior = ((unsigned)(tX - 1) < 6u) && ((unsigned)(tY - 1) < 6u);

#ifdef HAVE_TDM
    if (interior) {
        // TDM ignores EXEC: issue from exactly one wave via a *scalar* branch.
        unsigned wid = __builtin_amdgcn_readfirstlane((unsigned)tid) >> 5;
        if (wid == 0u) {
            tdm_load_tile(xin + (size_t)(row0 - 1) * IMG + (col0 - 1), &tile[0]);
        }
    } else
#endif
    {
        // border tiles: per-lane async copies; OOB elements zero-filled (conv padding)
        for (int e = tid; e < HALO * HALO; e += TPB) {
            int lr = e / HALO;
            int lc = e - lr * HALO;
            int gr = row0 - 1 + lr;
            int gc = col0 - 1 + lc;
            if ((unsigned)gr < (unsigned)IMG && (unsigned)gc < (unsigned)IMG) {
                async_g2l_b32(xin + (size_t)gr * IMG + gc, &tile[e]);
            } else {
                tile[e] = 0.0f;
            }
        }
    }

    for (int i = tid; i < 64 * NBINS; i += TPB) hist[i] = 0.0f;

#ifdef HAVE_TDM
    if (interior) {
        __builtin_amdgcn_s_wait_tensorcnt(0);   // no-op for waves with TENSORcnt==0
    } else
#endif
    {
        wait_asynccnt0();
    }
    __syncthreads();

    // ---- per-thread: one pooled cell (8x8 px) shared by 4 threads, 2 rows each ----
    const int cell = tid >> 2;            // 0..63  (cy = cell>>3, cx = cell&7)
    const int sub  = tid & 3;
    const int cy   = cell >> 3;
    const int cx   = cell & 7;
    const int r0   = cy * POOL + sub * 2; // first of 2 pixel rows
    const int c0   = cx * POOL;
    const unsigned histBase = (unsigned)(uintptr_t)&hist[cell * NBINS];

    // sector boundaries b_k = k*pi/10 (k=1..9)
    const float BCOS[9] = { 0.95105652f,  0.80901699f,  0.58778525f,  0.30901699f,
                            0.0f,        -0.30901699f, -0.58778525f, -0.80901699f,
                           -0.95105652f };
    const float BSIN[9] = { 0.30901699f,  0.58778525f,  0.80901699f,  0.95105652f,
                            1.0f,         0.95105652f,  0.80901699f,  0.58778525f,
                            0.30901699f };

    // sliding 4-row x 3-col window over the 8-wide run
    float w[4][3];
#pragma unroll
    for (int dr = 0; dr < 4; ++dr) {
        w[dr][0] = tile[(r0 + dr) * HALO + c0 + 0];
        w[dr][1] = tile[(r0 + dr) * HALO + c0 + 1];
    }
#pragma unroll
    for (int j = 0; j < POOL; ++j) {
#pragma unroll
        for (int dr = 0; dr < 4; ++dr)
            w[dr][2] = tile[(r0 + dr) * HALO + c0 + j + 2];
#pragma unroll
        for (int rr = 0; rr < 2; ++rr) {
            float a00 = w[rr][0],   a01 = w[rr][1],   a02 = w[rr][2];
            float a10 = w[rr+1][0],                   a12 = w[rr+1][2];
            float a20 = w[rr+2][0], a21 = w[rr+2][1], a22 = w[rr+2][2];
            // cross-correlation with sobel (gx) and sobel^T (gy), matching XLA conv
            float gxv = (a00 - a02) + 2.0f * (a10 - a12) + (a20 - a22);
            float gyv = (a00 + 2.0f * a01 + a02) - (a20 + 2.0f * a21 + a22);
            float mag = __builtin_sqrtf(gxv * gxv + gyv * gyv);
            // orientation is mod pi: normalize so sin(theta) >= 0
            bool flip = (gxv < 0.0f) || (gxv == 0.0f && gyv < 0.0f);
            float sx = flip ? -gxv : gxv;   // ~ sin(theta) * r
            float sy = flip ? -gyv : gyv;   // ~ cos(theta) * r
            // bin = #{k : sx*cos_k - sy*sin_k >= 0} = 9 - #negatives (sign-bit sum)
            int b = 9;
#pragma unroll
            for (int k = 0; k < 9; ++k) {
                float u = sx * BCOS[k] - sy * BSIN[k];
                b += __float_as_int(u) >> 31;      // -1 if negative
            }
            ds_fadd(histBase + (unsigned)b * 4u, mag);
        }
#pragma unroll
        for (int dr = 0; dr < 4; ++dr) {
            w[dr][0] = w[dr][1];
            w[dr][1] = w[dr][2];
        }
    }

    __syncthreads();

    // ---- write 640 pooled means: out[n][bin][py][px] ----
    const float inv = 1.0f / (float)(POOL * POOL);
    for (int i = tid; i < 64 * NBINS; i += TPB) {
        int c  = i / NBINS;
        int b  = i - c * NBINS;
        int py = tY * POOL + (c >> 3);
        int px = tX * POOL + (c & 7);
        out[(((size_t)n * NBINS + b) * (IMG / POOL) + py) * (IMG / POOL) + px]
            = hist[i] * inv;
    }
}

// ---------------- launch ----------------

extern "C" void kernel_launch(void* const* d_in, const int* in_sizes, int n_in,
                              void* d_out, int out_size, void* d_ws, size_t ws_size,
                              hipStream_t stream) {
    const float* x  = (const float*)d_in[0];
    float*      out = (float*)d_out;
    int nbatch = in_sizes[0] / (IMG * IMG);   // 32
    dim3 grid(IMG / TILE, IMG / TILE, nbatch); // 8 x 8 x 32 = 2048 blocks
    hog_fused_kernel<<<grid, TPB, 0, stream>>>(x, out);
}